// Retriever_22041772163282
// MI455X (gfx1250) — compile-verified
//
#include <hip/hip_runtime.h>
#include <hip/hip_bf16.h>
#include <math.h>

typedef __attribute__((ext_vector_type(2))) float v2f;
typedef __attribute__((ext_vector_type(8))) float v8f;
typedef __attribute__((ext_vector_type(4))) unsigned int u32x4;
typedef __attribute__((ext_vector_type(4))) int i32x4;
typedef __attribute__((ext_vector_type(8))) int i32x8;

#define NQ 32          // queries
#define TK 10          // top-k
#define DIM 128        // embedding dim
#define TP 132         // corpus tile pitch in dwords (TDM pad: 128 + 4) -> conflict-free B reads
#define SP 35          // score buffer pitch [col][q] -> conflict-free store+scan

__device__ __forceinline__ void topk_insert(float (&ts)[TK], int (&ti)[TK], float s, int idx) {
    if (s <= ts[TK - 1]) return;
    float cs = s; int ci = idx;
#pragma unroll
    for (int j = 0; j < TK; ++j) {
        if (cs > ts[j]) {
            float tv = ts[j]; int tvi = ti[j];
            ts[j] = cs; ti[j] = ci;
            cs = tv; ci = tvi;
        }
    }
}

// ---------------------------------------------------------------------------
// Kernel 1: normalize queries; emit k-pair-interleaved Qn^T:
//   half A (q 0..15):  qt[(kp)*32 + 2*q + dlo]        , d = 2*kp + dlo
//   half B (q 16..31): qt[2048 + (kp)*32 + 2*(q-16) + dlo]
// Layout matches LDS exactly so kernel 2 copies linearly.
// ---------------------------------------------------------------------------
__global__ __launch_bounds__(256) void normq_kernel(const float* __restrict__ Q,
                                                    float* __restrict__ qtOut) {
    __shared__ float part[256];
    __shared__ float inv[NQ];
    const int t = threadIdx.x;
    const int q = t >> 3, p = t & 7;
    float s = 0.f;
#pragma unroll
    for (int j = 0; j < 16; ++j) {
        float v = Q[q * DIM + p * 16 + j];
        s += v * v;
    }
    part[t] = s;
    __syncthreads();
    if (t < NQ) {
        float ss = 0.f;
#pragma unroll
        for (int j = 0; j < 8; ++j) ss += part[t * 8 + j];
        inv[t] = 1.0f / fmaxf(sqrtf(ss), 1e-12f);
    }
    __syncthreads();
#pragma unroll
    for (int j = 0; j < 16; ++j) {
        int idx = j * 256 + t;                 // 0..4095
        int half = idx >> 11;
        int o = idx & 2047;
        int kp = o >> 5, r = o & 31;
        int qq = half * 16 + (r >> 1);
        int d = 2 * kp + (r & 1);
        qtOut[idx] = Q[qq * DIM + d] * inv[qq];
    }
}

// ---------------------------------------------------------------------------
// Kernel 2: TDM-staged corpus stream + WMMA cosine scores + per-block top-k
// 128 threads = 4 wave32; each wave owns a 16-row tile per 64-row step.
// ---------------------------------------------------------------------------
__global__ __launch_bounds__(128) void score_topk_kernel(const float* __restrict__ corpus,
                                                         const float* __restrict__ qt,
                                                         float* __restrict__ candS,
                                                         int* __restrict__ candI,
                                                         int N, int RPB) {
    __shared__ float smem[14848];                 // 59392 B
    float* qtS     = smem;                        // 4096 (two 2048-dword halves)
    float* tileS   = smem + 4096;                 // 4 * 16*132 = 8448
    float* scoresS = smem + 4096 + 8448;          // 64*35 = 2240
    float* rnS     = smem + 4096 + 8448 + 2240;   // 64

    const int t    = threadIdx.x;
    const int lane = t & 31;
    const int wu   = __builtin_amdgcn_readfirstlane(t >> 5);  // wave id, forced SGPR
    const int h    = lane >> 4;                   // lane half -> K offset
    const int n16  = lane & 15;

    const long long blockStart = (long long)blockIdx.x * RPB;
    long long blockEnd = blockStart + RPB;
    if (blockEnd > N) blockEnd = N;

    // stage interleaved Qn^T (LDS layout == global layout)
#pragma unroll
    for (int j = 0; j < 32; ++j) {
        int idx = j * 128 + t;
        qtS[idx] = qt[idx];
    }
    __syncthreads();

    float ts[TK]; int ti[TK];
#pragma unroll
    for (int j = 0; j < TK; ++j) { ts[j] = -__builtin_inff(); ti[j] = -1; }

    float* myTile = tileS + wu * (16 * TP);
    const unsigned ldsOff = (unsigned)(unsigned long long)(uintptr_t)(void*)myTile;

    for (long long row0 = blockStart; row0 < blockEnd; row0 += 64) {
        const long long wrow0 = row0 + (long long)wu * 16;
        long long remll = (long long)N - wrow0;
        unsigned rem = remll > 0 ? (unsigned)remll : 0u;      // rows visible from tile start
        unsigned long long ga = (unsigned long long)(uintptr_t)(corpus + wrow0 * DIM);

        // ---- Tensor DMA descriptor: 2D tile 128x16 f32 -> LDS, pitch 132 dwords ----
        u32x4 g0; i32x8 g1;
        i32x4 gz4 = {0, 0, 0, 0};
        i32x8 gz8 = {0, 0, 0, 0, 0, 0, 0, 0};
        g0[0] = 1u;                                           // count=1, user mode
        g0[1] = ldsOff;                                       // lds_addr (bytes)
        g0[2] = (unsigned)ga;                                 // global_addr[31:0]
        g0[3] = (unsigned)((ga >> 32) & 0x01FFFFFFu) | (2u << 30);  // addr[56:32] | type=2
        g1[0] = (int)((2u << 16)        // data_size = 4B
                    | (1u << 20)        // pad_enable
                    | (6u << 22)        // pad_interval: every 128 dwords
                    | (3u << 25));      // pad_amount: 4 dwords  -> row pitch 132
        g1[1] = (int)(128u << 16);                            // tensor_dim0 = 128 (lo16)
        g1[2] = (int)((rem & 0xFFFFu) << 16);                 // tensor_dim1 lo16
        g1[3] = (int)(((rem >> 16) & 0xFFFFu) | (128u << 16)); // dim1 hi16 | tile_dim0=128
        g1[4] = 16;                                           // tile_dim1 = 16 rows
        g1[5] = 128;                                          // tensor_dim0_stride = 128
        g1[6] = 0;
        g1[7] = 0;

        asm volatile("" ::: "memory");
        __builtin_amdgcn_tensor_load_to_lds(g0, g1, gz4, gz4, gz8, 0);
        __builtin_amdgcn_s_wait_tensorcnt(0);
        asm volatile("" ::: "memory");

        // ---- row norms from LDS (one row per iteration, full-wave reduce) ----
#pragma unroll 4
        for (int j = 0; j < 16; ++j) {
            float4 v = *(const float4*)&myTile[j * TP + lane * 4];
            float s2 = v.x * v.x + v.y * v.y + v.z * v.z + v.w * v.w;
            s2 += __shfl_xor(s2, 1, 32);
            s2 += __shfl_xor(s2, 2, 32);
            s2 += __shfl_xor(s2, 4, 32);
            s2 += __shfl_xor(s2, 8, 32);
            s2 += __shfl_xor(s2, 16, 32);
            if (lane == 0) rnS[wu * 16 + j] = s2;
        }
        __builtin_amdgcn_wave_barrier();

        // ---- full-K WMMA: 32 k-steps x 2 query M-tiles ----
        v8f acc0 = {}; v8f acc1 = {};
#pragma unroll
        for (int s = 0; s < 32; ++s) {
            const int kp = 2 * s + h;             // k-pair index for this lane half
            v2f a0 = *(const v2f*)&qtS[kp * 32 + 2 * n16];          // queries 0..15
            v2f a1 = *(const v2f*)&qtS[2048 + kp * 32 + 2 * n16];   // queries 16..31
            v2f bb = *(const v2f*)&myTile[n16 * TP + 4 * s + 2 * h];
            acc0 = __builtin_amdgcn_wmma_f32_16x16x4_f32(
                false, a0, false, bb, (short)0, acc0, false, false);
            acc1 = __builtin_amdgcn_wmma_f32_16x16x4_f32(
                false, a1, false, bb, (short)0, acc1, false, false);
        }

        // ---- normalized scores -> LDS [col][q] ----
        __builtin_amdgcn_wave_barrier();
        const float rinv = 1.0f / fmaxf(sqrtf(rnS[wu * 16 + n16]), 1e-12f);
#pragma unroll
        for (int i = 0; i < 8; ++i) {
            const int q0 = i + 8 * h;
            scoresS[(wu * 16 + n16) * SP + q0]      = acc0[i] * rinv;
            scoresS[(wu * 16 + n16) * SP + 16 + q0] = acc1[i] * rinv;
        }
        __syncthreads();

        // ---- scan: thread owns (query t&31, 16-col chunk t>>5), private top-k ----
        {
            const int q = t & 31;
            const int c = t >> 5;                 // 0..3
            const long long gbase = row0 + (long long)c * 16;
#pragma unroll
            for (int j = 0; j < 16; ++j) {
                long long g = gbase + j;
                if (g < blockEnd) {
                    float s = scoresS[(c * 16 + j) * SP + q];
                    topk_insert(ts, ti, s, (int)g);
                }
            }
        }
        __syncthreads();
    }

    // ---- block merge: 4 lists per query -> 10 candidates ----
    float* ms = smem;
    int*   mi = (int*)(smem + 1280);
    __syncthreads();
#pragma unroll
    for (int j = 0; j < TK; ++j) { ms[t * TK + j] = ts[j]; mi[t * TK + j] = ti[j]; }
    __syncthreads();
    if (t < NQ) {
        float fs[TK]; int fi[TK];
#pragma unroll
        for (int j = 0; j < TK; ++j) { fs[j] = ms[t * TK + j]; fi[j] = mi[t * TK + j]; }
        for (int g = 1; g < 4; ++g) {
            int src = g * 32 + t;
#pragma unroll
            for (int j = 0; j < TK; ++j) {
                float s = ms[src * TK + j];
                if (s <= fs[TK - 1]) break;       // lists sorted desc
                topk_insert(fs, fi, s, mi[src * TK + j]);
            }
        }
        const long long o = ((long long)blockIdx.x * NQ + t) * TK;
#pragma unroll
        for (int j = 0; j < TK; ++j) { candS[o + j] = fs[j]; candI[o + j] = fi[j]; }
    }
}

// ---------------------------------------------------------------------------
// Kernel 3: merge per-block candidates -> final top-10 per query
// ---------------------------------------------------------------------------
__global__ __launch_bounds__(256) void merge_kernel(const float* __restrict__ candS,
                                                    const int* __restrict__ candI,
                                                    int nCand,   // NB * TK
                                                    float* __restrict__ outS,
                                                    int* __restrict__ outI) {
    __shared__ float ms[256 * TK];
    __shared__ int   mi[256 * TK];
    const int q = blockIdx.x;
    const int t = threadIdx.x;
    float ts[TK]; int ti[TK];
#pragma unroll
    for (int j = 0; j < TK; ++j) { ts[j] = -__builtin_inff(); ti[j] = -1; }

    for (int c = t; c < nCand; c += 256) {
        int blk = c / TK;
        int j = c - blk * TK;
        long long idx = (long long)blk * (NQ * TK) + q * TK + j;
        topk_insert(ts, ti, candS[idx], candI[idx]);
    }
#pragma unroll
    for (int j = 0; j < TK; ++j) { ms[t * TK + j] = ts[j]; mi[t * TK + j] = ti[j]; }
    __syncthreads();

    for (int stride = 128; stride >= 1; stride >>= 1) {
        if (t < stride) {
            int src = t + stride;
#pragma unroll
            for (int j = 0; j < TK; ++j) {
                float s = ms[src * TK + j];
                if (s <= ts[TK - 1]) break;
                topk_insert(ts, ti, s, mi[src * TK + j]);
            }
#pragma unroll
            for (int j = 0; j < TK; ++j) { ms[t * TK + j] = ts[j]; mi[t * TK + j] = ti[j]; }
        }
        __syncthreads();
    }
    if (t == 0) {
#pragma unroll
        for (int j = 0; j < TK; ++j) {
            outS[q * TK + j] = ts[j];
            outI[q * TK + j] = ti[j];
        }
    }
}

// ---------------------------------------------------------------------------
extern "C" void kernel_launch(void* const* d_in, const int* in_sizes, int n_in,
                              void* d_out, int out_size, void* d_ws, size_t ws_size,
                              hipStream_t stream) {
    const float* queries = (const float*)d_in[0];
    const float* corpus  = (const float*)d_in[1];
    const int N = in_sizes[1] / DIM;              // 1,000,000

    float* ws = (float*)d_ws;
    float* qt = ws;                               // 4096 floats: interleaved Qn^T

    size_t avail = (ws_size > (size_t)4096 * 4) ? ws_size - (size_t)4096 * 4 : 0;
    int NB = (int)(avail / (NQ * TK * 8));        // 2560 B of candidates per block
    if (NB > 2048) NB = 2048;
    if (NB < 1) NB = 1;
    float* candS = ws + 4096;
    int*   candI = (int*)(candS + (size_t)NB * NQ * TK);

    const int RPB = (N + NB - 1) / NB;            // rows per block

    normq_kernel<<<1, 256, 0, stream>>>(queries, qt);
    score_topk_kernel<<<NB, 128, 0, stream>>>(corpus, qt, candS, candI, N, RPB);
    // outputs: scores [32,10] f32, then indices [32,10] i32 (flat concat)
    merge_kernel<<<NQ, 256, 0, stream>>>(candS, candI, NB * TK,
                                         (float*)d_out, ((int*)d_out) + NQ * TK);
}